// BOW_11982958756127
// MI455X (gfx1250) — compile-verified
//
#include <hip/hip_runtime.h>
#include <hip/hip_bf16.h>

// Problem constants (match reference)
#define VOCAB 50000
#define BATCH 256
#define SEQ   512
#define M1    1024
#define M2    512
#define EMB   256

typedef __attribute__((ext_vector_type(2))) float v2f;
typedef __attribute__((ext_vector_type(8))) float v8f;

// ---------------------------------------------------------------------------
// Kernel 0: coalesced transpose W1 [M1, VOCAB] -> W1T [VOCAB, M1]
// 32x32 tiles via LDS; block (32,8), each thread moves 4 elements.
// ---------------------------------------------------------------------------
__global__ void transpose_w1_kernel(const float* __restrict__ W1,
                                    float* __restrict__ W1T) {
    __shared__ float tile[32][33];
    const int vBase = blockIdx.x * 32;   // vocab dim
    const int mBase = blockIdx.y * 32;   // M1 dim
    const int tx = threadIdx.x;          // 0..31
    const int ty = threadIdx.y;          // 0..7

#pragma unroll
    for (int i = 0; i < 4; ++i) {
        const int m = mBase + ty + i * 8;
        const int v = vBase + tx;
        float val = 0.0f;
        if (v < VOCAB) val = W1[(size_t)m * VOCAB + v];   // coalesced read
        tile[ty + i * 8][tx] = val;
    }
    __syncthreads();
#pragma unroll
    for (int i = 0; i < 4; ++i) {
        const int v = vBase + ty + i * 8;
        const int m = mBase + tx;
        if (v < VOCAB) W1T[(size_t)v * M1 + m] = tile[tx][ty + i * 8];  // coalesced write
    }
}

// ---------------------------------------------------------------------------
// Kernel 1: implicit-BOW layer 1.
// h1[b, :] = relu(b1 + sum_j W1T[idx[b,j], :])
// One block per batch row; 256 threads; thread t accumulates float4 covering
// m = 4t..4t+3. Each indexed row is a contiguous, fully coalesced 4KB read.
// ---------------------------------------------------------------------------
__global__ void bow_layer1_kernel(const int* __restrict__ idx,
                                  const float* __restrict__ W1T,
                                  const float* __restrict__ b1,
                                  float* __restrict__ h1) {
    const int b = blockIdx.x;
    const int t = threadIdx.x;  // 0..255
    __shared__ int sIdx[SEQ];
    sIdx[t]       = idx[b * SEQ + t];
    sIdx[t + 256] = idx[b * SEQ + 256 + t];
    __syncthreads();

    float4 acc = make_float4(0.f, 0.f, 0.f, 0.f);
    for (int j = 0; j < SEQ; j += 4) {
        const float4 x0 = ((const float4*)(W1T + (size_t)sIdx[j + 0] * M1))[t];
        const float4 x1 = ((const float4*)(W1T + (size_t)sIdx[j + 1] * M1))[t];
        const float4 x2 = ((const float4*)(W1T + (size_t)sIdx[j + 2] * M1))[t];
        const float4 x3 = ((const float4*)(W1T + (size_t)sIdx[j + 3] * M1))[t];
        if (j + 4 < SEQ) {
            // pull the next indexed row segment toward the caches (global_prefetch_b8)
            __builtin_prefetch(W1T + (size_t)sIdx[j + 4] * M1 + 4 * t, 0, 1);
        }
        acc.x += x0.x + x1.x + x2.x + x3.x;
        acc.y += x0.y + x1.y + x2.y + x3.y;
        acc.z += x0.z + x1.z + x2.z + x3.z;
        acc.w += x0.w + x1.w + x2.w + x3.w;
    }
    const float4 bias = ((const float4*)b1)[t];
    float4 r;
    r.x = fmaxf(acc.x + bias.x, 0.f);
    r.y = fmaxf(acc.y + bias.y, 0.f);
    r.z = fmaxf(acc.z + bias.z, 0.f);
    r.w = fmaxf(acc.w + bias.w, 0.f);
    ((float4*)(h1 + (size_t)b * M1))[t] = r;
}

// ---------------------------------------------------------------------------
// Kernel 2/3: dense fp32 GEMM + bias + relu via V_WMMA_F32_16X16X4_F32.
// out[M,N] = relu(A[M,K] @ W[N,K]^T + bias[N])
// One 16x16 output tile per wave; K-loop in steps of 4.
// A fragment (16x4 f32, 2 VGPRs): lanes 0-15 hold K=k,k+1; lanes 16-31 K=k+2,k+3.
// B fragment (4x16 f32, 2 VGPRs): same K split, lane%16 selects N.
// C/D (16x16 f32, 8 VGPRs): VGPR r -> M=r (lanes 0-15) / M=8+r (lanes 16-31).
// ---------------------------------------------------------------------------
__global__ void gemm_relu_wmma_kernel(const float* __restrict__ A,
                                      const float* __restrict__ W,
                                      const float* __restrict__ bias,
                                      float* __restrict__ out,
                                      int M, int N, int K) {
    const int lane   = threadIdx.x & 31;
    const int wave   = threadIdx.x >> 5;
    const int wpb    = blockDim.x >> 5;
    const int tile   = blockIdx.x * wpb + wave;
    const int tilesN = N >> 4;
    const int m0 = (tile / tilesN) << 4;
    const int n0 = (tile % tilesN) << 4;
    if (m0 >= M) return;  // wave-uniform

    const int lm   = lane & 15;
    const int kOff = (lane >> 4) << 1;  // 0 for lanes 0-15, 2 for lanes 16-31
    const float* aRow = A + (size_t)(m0 + lm) * K + kOff;
    const float* bRow = W + (size_t)(n0 + lm) * K + kOff;

    v8f c = {};
    for (int k = 0; k < K; k += 4) {
        v2f a, b;
        a.x = aRow[k];
        a.y = aRow[k + 1];
        b.x = bRow[k];
        b.y = bRow[k + 1];
        // 8 args: (neg_a, A, neg_b, B, c_mod, C, reuse_a, reuse_b)
        c = __builtin_amdgcn_wmma_f32_16x16x4_f32(
                false, a, false, b, (short)0, c, false, false);
    }

    const float bb  = bias[n0 + lm];
    const int mRow  = m0 + ((lane >> 4) << 3);  // +8 for lanes 16-31
    float* o = out + (size_t)mRow * N + n0 + lm;
#pragma unroll
    for (int r = 0; r < 8; ++r) {
        o[(size_t)r * N] = fmaxf(c[r] + bb, 0.f);
    }
}

// ---------------------------------------------------------------------------
// Launch
// ---------------------------------------------------------------------------
extern "C" void kernel_launch(void* const* d_in, const int* in_sizes, int n_in,
                              void* d_out, int out_size, void* d_ws, size_t ws_size,
                              hipStream_t stream) {
    const int*   idx = (const int*)  d_in[0];
    const float* W1  = (const float*)d_in[1];
    const float* b1  = (const float*)d_in[2];
    const float* W2  = (const float*)d_in[3];
    const float* b2  = (const float*)d_in[4];
    const float* W3  = (const float*)d_in[5];
    const float* b3  = (const float*)d_in[6];
    float* out = (float*)d_out;

    // workspace layout
    char* ws = (char*)d_ws;
    float* W1T = (float*)ws;                                  // VOCAB*M1 floats (200 MB)
    float* h1  = (float*)(ws + (size_t)VOCAB * M1 * sizeof(float));  // B*M1
    float* h2  = h1 + (size_t)BATCH * M1;                            // B*M2

    // 0) transpose W1 -> W1T
    {
        dim3 grid((VOCAB + 31) / 32, M1 / 32);
        dim3 block(32, 8);
        transpose_w1_kernel<<<grid, block, 0, stream>>>(W1, W1T);
    }
    // 1) implicit BOW + layer1
    bow_layer1_kernel<<<dim3(BATCH), dim3(256), 0, stream>>>(idx, W1T, b1, h1);

    // 2) layer2: h2 = relu(h1 @ W2^T + b2)   [256,1024] x [512,1024]^T
    {
        const int tiles = (BATCH / 16) * (M2 / 16);  // 512 tiles
        gemm_relu_wmma_kernel<<<dim3(tiles / 8), dim3(256), 0, stream>>>(
            h1, W2, b2, h2, BATCH, M2, M1);
    }
    // 3) layer3: out = relu(h2 @ W3^T + b3)  [256,512] x [256,512]^T
    {
        const int tiles = (BATCH / 16) * (EMB / 16);  // 256 tiles
        gemm_relu_wmma_kernel<<<dim3(tiles / 8), dim3(256), 0, stream>>>(
            h2, W3, b3, out, BATCH, EMB, M2);
    }
}